// VNInvAttentionLayer_89687507075986
// MI455X (gfx1250) — compile-verified
//
#include <hip/hip_runtime.h>
#include <math.h>

// ---------------- problem constants ----------------
constexpr int Bb = 16, Ll = 512, DM = 512, Hh = 8, Dd = 64;
constexpr long ROWS   = (long)Bb * Ll * 3;          // 24576 GEMM rows for projections
constexpr long QSZ    = (long)Bb * Ll * 3 * DM;     // 12,582,912 floats per q/k/v buffer
constexpr long BVH    = (long)Bb * 3 * Hh;          // 384 attention batches

typedef float v2f __attribute__((ext_vector_type(2)));
typedef float v8f __attribute__((ext_vector_type(8)));

// =====================================================================
// Generic batched WMMA GEMM (fp32, V_WMMA_F32_16X16X4_F32).
// One wave per block; each wave owns a 32(M) x 64(N) strip of C
// (2 A-fragments x 4 B-fragments = 8 accumulators), with register
// double-buffering of the next K-step so global-load latency overlaps
// the WMMAs (partial s_wait_loadcnt instead of wait-0 per step).
// Batch z decomposes as (b, v, h) with independent strides per matrix,
// so one kernel serves: projections (batch=1), Q@K^T (TRANSB),
// attn@V, and the output projection.
// Requires: M%32==0, N%64==0, K%4==0, full 32-lane waves (EXEC all 1s).
// =====================================================================
template<bool TRANSB>
__global__ void __launch_bounds__(32)
wmma_gemm(const float* __restrict__ A, const float* __restrict__ B,
          float* __restrict__ C,
          int K, int lda, int ldb, int ldc,
          long sAb, long sAv, long sAh,
          long sBb, long sBv, long sBh,
          long sCb, long sCv, long sCh,
          float scale)
{
    const int z  = blockIdx.z;
    const int b  = z / (3 * Hh);
    const int v  = (z / Hh) % 3;
    const int h  = z % Hh;
    const float* Ap = A + (long)b * sAb + (long)v * sAv + (long)h * sAh;
    const float* Bp = B + (long)b * sBb + (long)v * sBv + (long)h * sBh;
    float*       Cp = C + (long)b * sCb + (long)v * sCv + (long)h * sCh;

    const int lane = threadIdx.x & 31;
    const int lm   = lane & 15;         // position within 16
    const int hi   = lane >> 4;         // 0: K pair {0,1}, 1: K pair {2,3}
    const int m0   = blockIdx.y * 32;
    const int n0   = blockIdx.x * 64;

    v8f acc[2][4] = {};                 // eight 16x16 fp32 accumulators

    // A panel pointers: rows m0+lm and m0+16+lm, starting at K pair 2*hi
    const float* pa0 = Ap + (long)(m0 + lm) * lda + 2 * hi;
    const float* pa1 = pa0 + (long)16 * lda;

    // B pointers
    const float* pbT[4];                // TRANSB: one row pointer per N tile
    const float* pbN = nullptr;         // !TRANSB: one base, tiles via +16*t
    if constexpr (TRANSB) {
#pragma unroll
        for (int t = 0; t < 4; ++t)
            pbT[t] = Bp + (long)(n0 + 16 * t + lm) * ldb + 2 * hi;
    } else {
        pbN = Bp + (long)(2 * hi) * ldb + n0 + lm;
    }

    struct Frag { v2f a0, a1, bf[4]; };

    auto load_step = [&](Frag& f) {
        f.a0.x = pa0[0]; f.a0.y = pa0[1];
        f.a1.x = pa1[0]; f.a1.y = pa1[1];
        pa0 += 4; pa1 += 4;
        if constexpr (TRANSB) {         // B[k][n] = Bp[n*ldb + k]
#pragma unroll
            for (int t = 0; t < 4; ++t) {
                f.bf[t].x = pbT[t][0];
                f.bf[t].y = pbT[t][1];
                pbT[t] += 4;
            }
        } else {                        // B[k][n] = Bp[k*ldb + n]
#pragma unroll
            for (int t = 0; t < 4; ++t) {
                f.bf[t].x = pbN[16 * t];
                f.bf[t].y = pbN[16 * t + ldb];
            }
            pbN += 4 * (long)ldb;
        }
    };

    auto compute = [&](const Frag& f) {
#pragma unroll
        for (int t = 0; t < 4; ++t) {
            acc[0][t] = __builtin_amdgcn_wmma_f32_16x16x4_f32(
                            false, f.a0, false, f.bf[t], (short)0,
                            acc[0][t], false, false);
            acc[1][t] = __builtin_amdgcn_wmma_f32_16x16x4_f32(
                            false, f.a1, false, f.bf[t], (short)0,
                            acc[1][t], false, false);
        }
    };

    Frag cur, nxt;
    load_step(cur);
    for (int k = 4; k < K; k += 4) {
        __builtin_prefetch(pa0 + 16, 0, 0);     // global_prefetch_b8: A panel ahead
        load_step(nxt);                         // issue next-step loads first
        compute(cur);                           // overlap WMMAs with those loads
        cur = nxt;
    }
    compute(cur);

    // C/D layout: VGPR r -> row r (lanes 0-15) / row r+8 (lanes 16-31)
#pragma unroll
    for (int mi = 0; mi < 2; ++mi) {
#pragma unroll
        for (int t = 0; t < 4; ++t) {
            const int n = n0 + 16 * t + lm;
#pragma unroll
            for (int r = 0; r < 8; ++r) {
                const int m = m0 + 16 * mi + r + 8 * hi;
                Cp[(long)m * ldc + n] = acc[mi][t][r] * scale;
            }
        }
    }
}

// =====================================================================
// VN std-feature: per (b,l,h) -> rotation-invariant features, in place.
// Block = one (b,l); 8 waves = 8 heads. Lane j owns all 3 SO(3)
// components of output channel j, so the reflect dot/d^2 reductions of
// vn_leaky_relu are lane-local (no cross-lane reduction needed).
// =====================================================================
__global__ void __launch_bounds__(256)
vn_feat(float* __restrict__ X,          // (B,L,3,H,D) in place
        const float* __restrict__ W1,   // (64,32) row-major
        const float* __restrict__ W1d,  // (64,32)
        const float* __restrict__ W2,   // (32,16)
        const float* __restrict__ W2d,  // (32,16)
        const float* __restrict__ Wz)   // (16,3)
{
    __shared__ float xs [Hh][3][64];
    __shared__ float z1s[Hh][3][32];
    __shared__ float z2s[Hh][3][16];
    __shared__ float z3s[Hh][3][3];

    const int h    = threadIdx.x >> 5;
    const int lane = threadIdx.x & 31;
    float* base = X + (long)blockIdx.x * (3 * Hh * Dd);   // (b*L+l)*1536

    // stage x tile (3 x 64 per head): 2 channels per lane
#pragma unroll
    for (int v = 0; v < 3; ++v) {
        xs[h][v][2 * lane]     = base[v * 512 + h * 64 + 2 * lane];
        xs[h][v][2 * lane + 1] = base[v * 512 + h * 64 + 2 * lane + 1];
    }
    __syncthreads();

    // ---- layer 1: D=64 -> 32, lane j = output channel j ----
    {
        float p[3], dv[3], dot = 0.f, dsq = 0.f;
#pragma unroll
        for (int v = 0; v < 3; ++v) {
            float pp = 0.f, dd = 0.f;
            for (int c = 0; c < 64; ++c) {
                const float xv = xs[h][v][c];
                pp += xv * W1 [c * 32 + lane];
                dd += xv * W1d[c * 32 + lane];
            }
            p[v] = pp; dv[v] = dd;
            dot += pp * dd; dsq += dd * dd;
        }
        const float f = dot / (dsq + 1e-6f);
#pragma unroll
        for (int v = 0; v < 3; ++v) {
            const float pr = (dot >= 0.f) ? p[v] : (p[v] - f * dv[v]);
            z1s[h][v][lane] = 0.2f * p[v] + 0.8f * pr;
        }
    }
    __syncthreads();

    // ---- layer 2: 32 -> 16, lanes 0..15 ----
    if (lane < 16) {
        float p[3], dv[3], dot = 0.f, dsq = 0.f;
#pragma unroll
        for (int v = 0; v < 3; ++v) {
            float pp = 0.f, dd = 0.f;
            for (int c = 0; c < 32; ++c) {
                const float xv = z1s[h][v][c];
                pp += xv * W2 [c * 16 + lane];
                dd += xv * W2d[c * 16 + lane];
            }
            p[v] = pp; dv[v] = dd;
            dot += pp * dd; dsq += dd * dd;
        }
        const float f = dot / (dsq + 1e-6f);
#pragma unroll
        for (int v = 0; v < 3; ++v) {
            const float pr = (dot >= 0.f) ? p[v] : (p[v] - f * dv[v]);
            z2s[h][v][lane] = 0.2f * p[v] + 0.8f * pr;
        }
    }
    __syncthreads();

    // ---- learned frame: z3 = z2 @ Wz  (3x3), lanes 0..8 ----
    if (lane < 9) {
        const int v = lane / 3, kk = lane % 3;
        float s = 0.f;
        for (int c = 0; c < 16; ++c) s += z2s[h][v][c] * Wz[c * 3 + kk];
        z3s[h][v][kk] = s;
    }
    __syncthreads();

    // ---- invariant contraction: out[k][c] = sum_v x[v][c] * z3[v][k] ----
#pragma unroll
    for (int k = 0; k < 3; ++k) {
#pragma unroll
        for (int u = 0; u < 2; ++u) {
            const int c = 2 * lane + u;
            const float s = xs[h][0][c] * z3s[h][0][k]
                          + xs[h][1][c] * z3s[h][1][k]
                          + xs[h][2][c] * z3s[h][2][k];
            base[k * 512 + h * 64 + c] = s;
        }
    }
}

// =====================================================================
// Row softmax, in place. One wave32 per 512-wide row; 16 elems/lane,
// shuffle reductions for max and sum.
// =====================================================================
__global__ void __launch_bounds__(256)
softmax_rows(float* __restrict__ attn, long nrows)
{
    const int  lane = threadIdx.x & 31;
    const long row  = (long)blockIdx.x * (blockDim.x >> 5) + (threadIdx.x >> 5);
    if (row >= nrows) return;
    float* p = attn + row * 512;

    float v[16];
    float m = -3.402823466e38f;
#pragma unroll
    for (int i = 0; i < 16; ++i) { v[i] = p[i * 32 + lane]; m = fmaxf(m, v[i]); }
#pragma unroll
    for (int o = 16; o > 0; o >>= 1) m = fmaxf(m, __shfl_xor(m, o, 32));

    float s = 0.f;
#pragma unroll
    for (int i = 0; i < 16; ++i) { v[i] = __expf(v[i] - m); s += v[i]; }
#pragma unroll
    for (int o = 16; o > 0; o >>= 1) s += __shfl_xor(s, o, 32);

    const float inv = 1.0f / s;
#pragma unroll
    for (int i = 0; i < 16; ++i) p[i * 32 + lane] = v[i] * inv;
}

// =====================================================================
extern "C" void kernel_launch(void* const* d_in, const int* in_sizes, int n_in,
                              void* d_out, int out_size, void* d_ws, size_t ws_size,
                              hipStream_t stream)
{
    const float* queries = (const float*)d_in[0];
    const float* keys    = (const float*)d_in[1];
    const float* values  = (const float*)d_in[2];
    const float* Wq      = (const float*)d_in[3];
    const float* Wk      = (const float*)d_in[4];
    const float* Wv      = (const float*)d_in[5];
    const float* Wout    = (const float*)d_in[6];
    // per-path VN weights: W1, W1d, W2, W2d, Wz
    const float* qW[5] = {(const float*)d_in[7],  (const float*)d_in[8],
                          (const float*)d_in[9],  (const float*)d_in[10],
                          (const float*)d_in[11]};
    const float* kW[5] = {(const float*)d_in[12], (const float*)d_in[13],
                          (const float*)d_in[14], (const float*)d_in[15],
                          (const float*)d_in[16]};
    const float* vW[5] = {(const float*)d_in[17], (const float*)d_in[18],
                          (const float*)d_in[19], (const float*)d_in[20],
                          (const float*)d_in[21]};

    float* ws   = (float*)d_ws;
    float* qb   = ws;                // (B,L,3,H,D)  -> becomes q_inv in place
    float* kb   = ws + QSZ;
    float* vb   = ws + 2 * QSZ;
    float* outp = ws + 3 * QSZ;      // pre-projection attention output

    float* out0 = (float*)d_out;     // (B,L,3,DM)
    float* attn = out0 + QSZ;        // (B,3,H,L,S)

    const dim3 blk(32);
    const dim3 gProj(DM / 64, (int)(ROWS / 32), 1);          // (8, 768)

    // ---- 1) q/k/v projections: fp32 WMMA GEMMs (24576x512 @ 512x512) ----
    wmma_gemm<false><<<gProj, blk, 0, stream>>>(queries, Wq, qb,
        512, 512, 512, 512, 0,0,0, 0,0,0, 0,0,0, 1.0f);
    wmma_gemm<false><<<gProj, blk, 0, stream>>>(keys,    Wk, kb,
        512, 512, 512, 512, 0,0,0, 0,0,0, 0,0,0, 1.0f);
    wmma_gemm<false><<<gProj, blk, 0, stream>>>(values,  Wv, vb,
        512, 512, 512, 512, 0,0,0, 0,0,0, 0,0,0, 1.0f);

    // ---- 2) rotation-invariant VN features (in place) ----
    vn_feat<<<Bb * Ll, 256, 0, stream>>>(qb, qW[0], qW[1], qW[2], qW[3], qW[4]);
    vn_feat<<<Bb * Ll, 256, 0, stream>>>(kb, kW[0], kW[1], kW[2], kW[3], kW[4]);
    vn_feat<<<Bb * Ll, 256, 0, stream>>>(vb, vW[0], vW[1], vW[2], vW[3], vW[4]);

    // ---- 3) scores: attn_raw[b,v,h,l,s] = (1/8) * Qi @ Ki^T  (K=64) ----
    {
        const long sQ_b = (long)Ll * 3 * Hh * Dd;  // 786432
        const long sQ_v = (long)Hh * Dd;           // 512
        const long sQ_h = (long)Dd;                // 64
        const long LL   = (long)Ll * Ll;           // 262144
        const dim3 g(Ll / 64, Ll / 32, (int)BVH);  // (8, 16, 384)
        wmma_gemm<true><<<g, blk, 0, stream>>>(qb, kb, attn,
            64, 3 * Hh * Dd, 3 * Hh * Dd, Ll,
            sQ_b, sQ_v, sQ_h,
            sQ_b, sQ_v, sQ_h,
            3 * Hh * LL, Hh * LL, LL,
            0.125f);
    }

    // ---- 4) softmax over s, in place on the attn output region ----
    softmax_rows<<<(int)(BVH * Ll / 8), 256, 0, stream>>>(attn, BVH * Ll);

    // ---- 5) O[b,l,v,h,d] = attn @ Vi  (512x64, K=512) ----
    {
        const long sV_b = (long)Ll * 3 * Hh * Dd;
        const long sV_v = (long)Hh * Dd;
        const long sV_h = (long)Dd;
        const long LL   = (long)Ll * Ll;
        const dim3 g(Dd / 64, Ll / 32, (int)BVH);  // (1, 16, 384)
        wmma_gemm<false><<<g, blk, 0, stream>>>(attn, vb, outp,
            512, Ll, 3 * Hh * Dd, 3 * Hh * Dd,
            3 * Hh * LL, Hh * LL, LL,
            sV_b, sV_v, sV_h,
            sV_b, sV_v, sV_h,
            1.0f);
    }

    // ---- 6) output projection: (24576x512) @ Wout -> d_out[0:QSZ] ----
    wmma_gemm<false><<<gProj, blk, 0, stream>>>(outp, Wout, out0,
        512, 512, 512, 512, 0,0,0, 0,0,0, 0,0,0, 1.0f);
}